// EucCluster_28845000360192
// MI455X (gfx1250) — compile-verified
//
#include <hip/hip_runtime.h>
#include <cstdint>

// ---------------- problem constants (fixed by reference) ----------------
#define N_PTS   200000   // pool size
#define M_CTR   512      // number of centers
#define DIM     128      // feature dim
#define ROWS_WG 64       // rows (points) per workgroup in the GEMM kernel
#define COL_CHUNK 32     // center columns staged in LDS at a time
#define NCHUNK  (M_CTR / COL_CHUNK)          // 16
#define CPAD    4        // LDS row pad (floats): keeps 16B align + no bank conflicts
#define SEGSZ   256      // row segment size for greedy candidates
#define NSEG    ((N_PTS + SEGSZ - 1) / SEGSZ) // 782

typedef float v2f __attribute__((ext_vector_type(2)));
typedef float v8f __attribute__((ext_vector_type(8)));
typedef unsigned long long u64;

// ---------------------------------------------------------------------------
// Kernel 0: init. c2[m] = |c_m|^2, candidate array = ~0, copy centers to out.
// Must run every launch: harness poisons ws once and never re-poisons.
// ---------------------------------------------------------------------------
__global__ void k_init(const float* __restrict__ c, float* __restrict__ c2,
                       u64* __restrict__ cand, float* __restrict__ out_centers) {
  int tid = blockIdx.x * blockDim.x + threadIdx.x;
  int nth = gridDim.x * blockDim.x;
  for (int i = tid; i < M_CTR * DIM / 4; i += nth)
    ((float4*)out_centers)[i] = ((const float4*)c)[i];
  for (int i = tid; i < M_CTR * NSEG; i += nth)
    cand[i] = ~0ull;
  for (int i = tid; i < M_CTR; i += nth) {
    float s = 0.f;
    for (int k = 0; k < DIM; ++k) { float v = c[i * DIM + k]; s += v * v; }
    c2[i] = s;
  }
}

// ---------------------------------------------------------------------------
// Kernel 1: fp32 WMMA GEMM (G = x . c^T) + distance epilogue.
//  - A fragments register-resident across all 16 col-chunks (1 global read of x).
//  - Center chunks double-buffered in LDS via global_load_async_to_lds_b128
//    (ASYNCcnt), overlapped with WMMA compute on the other buffer.
//  - Inner loop: one ds_load_b64 (B frag) per v_wmma_f32_16x16x4_f32.
//  grid: N_PTS/ROWS_WG = 3125 blocks x 256 threads (8 waves).
//  wave w -> C tile rows (w&3)*16, cols (w>>2)*16 of the 32-col chunk.
// ---------------------------------------------------------------------------
__launch_bounds__(256)
__global__ void k_dist(const float* __restrict__ x, const float* __restrict__ c,
                       const float* __restrict__ c2, float* __restrict__ x2g,
                       u64* __restrict__ cand, float* __restrict__ out_mind) {
  __shared__ __align__(16) float cs[2][COL_CHUNK][DIM + CPAD]; // [buf][col][k]
  __shared__ unsigned rowMin[ROWS_WG];   // float bits, monotone for dist >= 0
  __shared__ u64 colMin[M_CTR];          // packed (distbits<<32)|row
  __shared__ float x2s[ROWS_WG];

  const int t    = threadIdx.x;
  const int row0 = blockIdx.x * ROWS_WG;
  const int w    = t >> 5, lid = t & 31;
  const int r0   = (w & 3) * 16;        // row-tile base within WG
  const int c0   = (w >> 2) * 16;       // col-tile base within chunk
  const int half = (lid < 16) ? 0 : 2;  // A/B K-pair select per ISA layout
  const int mn   = lid & 15;            // M index for A, N index for B
  const int hi8  = (lid < 16) ? 0 : 8;

  // ---- kick async stage of center chunk 0 into cs[0] ----
  // chunk = 32 cols x 512B = 1024 x 16B pieces; 4 per thread.
  #pragma unroll
  for (int q = 0; q < 4; ++q) {
    int lin = t + q * 256;
    int r = lin >> 5;                   // col within chunk
    int j = lin & 31;                   // 16B piece within the 512B row
    unsigned ldsoff = (unsigned)(uintptr_t)&cs[0][r][0] + (unsigned)(j * 16);
    const char* g = (const char*)(c + (size_t)r * DIM) + j * 16;
    asm volatile("global_load_async_to_lds_b128 %0, %1, off"
                 :: "v"(ldsoff), "v"(g) : "memory");
  }

  // ---- A fragments: 64 floats/lane, register-resident for all chunks ----
  v2f a[32];
  const float* xrow = x + (size_t)(row0 + r0 + mn) * DIM + half;
  #pragma unroll
  for (int ks = 0; ks < 32; ++ks) a[ks] = *(const v2f*)(xrow + ks * 4);

  // row norms: lane l and l^16 hold the two K-halves of the same row
  float xx = 0.f;
  #pragma unroll
  for (int ks = 0; ks < 32; ++ks) xx += a[ks].x * a[ks].x + a[ks].y * a[ks].y;
  xx += __shfl_xor(xx, 16);
  if (w < 4 && lid < 16) { x2s[r0 + mn] = xx; x2g[row0 + r0 + mn] = xx; }

  if (t < ROWS_WG) rowMin[t] = 0xFFFFFFFFu;
  for (int i = t; i < M_CTR; i += 256) colMin[i] = ~0ull;

  unsigned rmin[8];
  #pragma unroll
  for (int j = 0; j < 8; ++j) rmin[j] = 0xFFFFFFFFu;

  for (int chunk = 0; chunk < NCHUNK; ++chunk) {
    const int b = chunk & 1;
    // current chunk's async copy complete + visible workgroup-wide
    asm volatile("s_wait_asynccnt 0x0" ::: "memory");
    __syncthreads();
    // stage next chunk into the other buffer; safe: its previous consumers
    // (chunk-1's reads) finished before the barrier above
    if (chunk + 1 < NCHUNK) {
      const int nb = b ^ 1;
      const size_t gbase = (size_t)(chunk + 1) * COL_CHUNK * DIM;
      #pragma unroll
      for (int q = 0; q < 4; ++q) {
        int lin = t + q * 256;
        int r = lin >> 5;
        int j = lin & 31;
        unsigned ldsoff = (unsigned)(uintptr_t)&cs[nb][r][0] + (unsigned)(j * 16);
        const char* g = (const char*)(c + gbase + (size_t)r * DIM) + j * 16;
        asm volatile("global_load_async_to_lds_b128 %0, %1, off"
                     :: "v"(ldsoff), "v"(g) : "memory");
      }
    }

    // two accumulators break the WMMA->WMMA RAW chain on C
    v8f acc0 = {}, acc1 = {};
    const float* bcol = &cs[b][c0 + mn][half];
    #pragma unroll
    for (int ks = 0; ks < 32; ks += 2) {
      v2f b0 = *(const v2f*)(bcol + ks * 4);
      v2f b1 = *(const v2f*)(bcol + ks * 4 + 4);
      acc0 = __builtin_amdgcn_wmma_f32_16x16x4_f32(
          false, a[ks],     false, b0, (short)0, acc0, false, false);
      acc1 = __builtin_amdgcn_wmma_f32_16x16x4_f32(
          false, a[ks + 1], false, b1, (short)0, acc1, false, false);
    }

    // epilogue: lane holds col = c0+mn, rows r0 + j (+8 for hi lanes)
    const int col = chunk * COL_CHUNK + c0 + mn;
    const float cc2 = c2[col];
    u64 colpk = ~0ull;
    #pragma unroll
    for (int j = 0; j < 8; ++j) {
      int ml = r0 + j + hi8;
      float g = acc0[j] + acc1[j];
      float d2 = x2s[ml] + cc2 - 2.0f * g;
      float dist = sqrtf(fmaxf(d2, 0.0f));
      unsigned bits = __float_as_uint(dist);
      rmin[j] = bits < rmin[j] ? bits : rmin[j];          // per-row, in regs
      u64 pk = ((u64)bits << 32) | (unsigned)(row0 + ml);
      colpk = pk < colpk ? pk : colpk;                    // per-col, in regs
    }
    atomicMin(&colMin[col], colpk);                       // 1 DS atomic/chunk
  }

  #pragma unroll
  for (int j = 0; j < 8; ++j) atomicMin(&rowMin[r0 + j + hi8], rmin[j]);
  __syncthreads();

  if (t < ROWS_WG) out_mind[row0 + t] = __uint_as_float(rowMin[t]);
  const int seg = row0 >> 8;   // 256-row segments (4 WGs share one segment)
  for (int i = t; i < M_CTR; i += 256)
    atomicMin(&cand[(u64)i * NSEG + seg], colMin[i]);
}

// ---------------------------------------------------------------------------
// Kernel 2: exact sequential greedy over 512 columns. One WG, 1024 threads.
// A stored candidate is exact for the unused set unless its row is used; only
// then recompute that 256-row segment from raw data, excluding used rows.
// ---------------------------------------------------------------------------
__launch_bounds__(1024)
__global__ void k_greedy(const float* __restrict__ x, const float* __restrict__ c,
                         const float* __restrict__ c2, const float* __restrict__ x2,
                         const u64* __restrict__ cand, float* __restrict__ out_idx) {
  __shared__ unsigned usedMask[(N_PTS + 31) / 32];  // 25 KB bitmask
  __shared__ u64 cands[NSEG];
  __shared__ int staleList[NSEG];
  __shared__ int staleCount;
  __shared__ u64 minSlot;
  __shared__ float cvec[DIM];

  const int t = threadIdx.x;
  for (int i = t; i < (N_PTS + 31) / 32; i += 1024) usedMask[i] = 0u;
  __syncthreads();

  for (int i = 0; i < M_CTR; ++i) {
    if (t < DIM) cvec[t] = c[i * DIM + t];
    if (t == 0) { staleCount = 0; minSlot = ~0ull; }
    __syncthreads();

    // load candidates; flag segments whose winner is already used
    for (int s = t; s < NSEG; s += 1024) {
      u64 pc = cand[(u64)i * NSEG + s];
      if (pc != ~0ull) {
        unsigned r = (unsigned)pc;
        if ((usedMask[r >> 5] >> (r & 31)) & 1u) {
          int slot = atomicAdd(&staleCount, 1);
          staleList[slot] = s;
          pc = ~0ull;                      // will be recomputed
        }
      }
      cands[s] = pc;
    }
    __syncthreads();

    const int nst = staleCount;
    const float ci2 = c2[i];
    for (int q = 0; q < nst; ++q) {        // rare: recompute stale segment
      int s  = staleList[q];
      int rl = t >> 2;                     // 256 rows x 4 lanes/row
      int part = t & 3;
      int r  = s * SEGSZ + rl;
      float s0 = 0.f;
      bool live = (r < N_PTS) && !((usedMask[r >> 5] >> (r & 31)) & 1u);
      if (live) {
        const float* xr = x + (size_t)r * DIM;
        for (int k = part * 32; k < part * 32 + 32; ++k) s0 += xr[k] * cvec[k];
      }
      s0 += __shfl_xor(s0, 1, 4);
      s0 += __shfl_xor(s0, 2, 4);
      if (live && part == 0) {
        float d2 = x2[r] + ci2 - 2.0f * s0;
        float dist = sqrtf(fmaxf(d2, 0.0f));
        u64 best = ((u64)__float_as_uint(dist) << 32) | (unsigned)r;
        atomicMin(&cands[s], best);
      }
      __syncthreads();
    }

    // global min over segments; packed compare = min dist, lowest row on ties
    for (int s = t; s < NSEG; s += 1024) atomicMin(&minSlot, cands[s]);
    __syncthreads();
    if (t == 0) {
      unsigned r = (unsigned)minSlot;
      out_idx[i] = (float)r;               // exact: r < 2^24
      usedMask[r >> 5] |= (1u << (r & 31));
    }
    __syncthreads();
  }
}

// ---------------------------------------------------------------------------
// Launch. d_out (float*): [0,512) indices, [512,512+200000) min dists,
// then 512*128 center copy. ws: c2 (2KB) | x2 (800KB) | cand (3.2MB).
// ---------------------------------------------------------------------------
extern "C" void kernel_launch(void* const* d_in, const int* in_sizes, int n_in,
                              void* d_out, int out_size, void* d_ws, size_t ws_size,
                              hipStream_t stream) {
  (void)in_sizes; (void)n_in; (void)out_size; (void)ws_size;
  const float* x = (const float*)d_in[0];
  const float* c = (const float*)d_in[1];

  float* out = (float*)d_out;
  float* out_idx     = out;
  float* out_mind    = out + M_CTR;
  float* out_centers = out + M_CTR + N_PTS;

  char* ws = (char*)d_ws;
  float* c2 = (float*)ws;
  float* x2 = (float*)(ws + M_CTR * sizeof(float));
  u64*  cand = (u64*)(ws + (size_t)(M_CTR + N_PTS) * sizeof(float)); // 8B aligned

  k_init<<<256, 256, 0, stream>>>(c, c2, cand, out_centers);
  k_dist<<<N_PTS / ROWS_WG, 256, 0, stream>>>(x, c, c2, x2, cand, out_mind);
  k_greedy<<<1, 1024, 0, stream>>>(x, c, c2, x2, cand, out_idx);
}